// IDLoss_79972291052000
// MI455X (gfx1250) — compile-verified
//
#include <hip/hip_runtime.h>
#include <hip/hip_bf16.h>

#define NROWS   200000
#define CCH     384
#define GRP     383
#define MM      384
#define EPSF    1e-5f
#define NTILES  12500          // NROWS / 16
#define NB1     600            // blocks for the streaming kernel
#define WAVES   6              // waves per block (192 threads, wave32)
#define LSTRIDE 68             // padded floats per staged row (68 % 64 == 4 -> conflict free)
#define SQRT_C  19.595917942265426f   // sqrt(384)

typedef float v2f __attribute__((ext_vector_type(2)));
typedef float v8f __attribute__((ext_vector_type(8)));
typedef int   v4i __attribute__((ext_vector_type(4)));

#if __has_builtin(__builtin_amdgcn_global_load_async_to_lds_b128)
#define HAVE_ASYNC 1
#else
#define HAVE_ASYNC 0
#endif

typedef __attribute__((address_space(1))) v4i gv4i_t;   // global (device) int4
typedef __attribute__((address_space(3))) v4i lv4i_t;   // LDS (shared) int4

template <int N>
__device__ __forceinline__ void wait_async_le() {
#if HAVE_ASYNC
#if __has_builtin(__builtin_amdgcn_s_wait_asynccnt)
    __builtin_amdgcn_s_wait_asynccnt(N);
#else
    asm volatile("s_wait_asynccnt %0" :: "n"(N) : "memory");
#endif
#endif
}

// Stage 16 rows x 64 cols of pred into an LDS tile.
// Per instruction: lanes 0-15 -> one 256B row segment, lanes 16-31 -> the next.
__device__ __forceinline__ void stage_load(float* __restrict__ dst,
                                           const float* __restrict__ src,
                                           int lhalf, int lmod) {
#if HAVE_ASYNC
    #pragma unroll
    for (int it = 0; it < 8; ++it) {
        const int r = it * 2 + lhalf;
        __builtin_amdgcn_global_load_async_to_lds_b128(
            (gv4i_t*)(src + (size_t)r * CCH + lmod * 4),
            (lv4i_t*)(dst + r * LSTRIDE + lmod * 4),
            0, 0);
    }
#else
    #pragma unroll
    for (int it = 0; it < 8; ++it) {
        const int r = it * 2 + lhalf;
        const float4 v = *(const float4*)(src + (size_t)r * CCH + lmod * 4);
        *(float4*)(dst + r * LSTRIDE + lmod * 4) = v;
    }
#endif
}

__device__ __forceinline__ float pick8(v8f v, int i) {
    float r01 = (i & 1) ? v[1] : v[0];
    float r23 = (i & 1) ? v[3] : v[2];
    float r45 = (i & 1) ? v[5] : v[4];
    float r67 = (i & 1) ? v[7] : v[6];
    float r03 = (i & 2) ? r23 : r01;
    float r47 = (i & 2) ? r67 : r45;
    return (i & 4) ? r47 : r03;
}

// ---------------------------------------------------------------------------
// Kernel 1: stream pred_id once; per-row sum and sum-of-squares via WMMA f32;
// async global->LDS staging (double buffered); scatter by target_id into
// per-block LDS tables; write per-block partials.
// ---------------------------------------------------------------------------
__global__ __launch_bounds__(192) void k1_group_stats(
    const float* __restrict__ pred, const int* __restrict__ tgt,
    float* __restrict__ psum, float* __restrict__ psq, float* __restrict__ pcnt)
{
    __shared__ float stage[WAVES][2][16 * LSTRIDE];
    __shared__ float lsum[GRP], lsq[GRP], lcnt[GRP];

    const int t = threadIdx.x;
    for (int g = t; g < GRP; g += 192) { lsum[g] = 0.f; lsq[g] = 0.f; lcnt[g] = 0.f; }
    __syncthreads();

    const int wave  = t >> 5;
    const int lane  = t & 31;
    const int lhalf = lane >> 4;   // 0: lanes 0-15, 1: lanes 16-31
    const int lmod  = lane & 15;

    const v2f ones = {1.0f, 1.0f};

    for (int tile = blockIdx.x * WAVES + wave; tile < NTILES; tile += NB1 * WAVES) {
        const int row0 = tile * 16;
        const float* base = pred + (size_t)row0 * CCH;
        v8f accS = {0.f,0.f,0.f,0.f,0.f,0.f,0.f,0.f};   // row sums (B = ones)
        v8f accQ = {0.f,0.f,0.f,0.f,0.f,0.f,0.f,0.f};   // Gram tile (diag = sumsq)

        stage_load(stage[wave][0], base, lhalf, lmod);

        #pragma unroll
        for (int s = 0; s < 6; ++s) {
            float* cur = stage[wave][s & 1];
            if (s < 5) {
                stage_load(stage[wave][(s + 1) & 1], base + (s + 1) * 64, lhalf, lmod);
                wait_async_le<8>();    // current stage's 8 copies complete
            } else {
                wait_async_le<0>();
            }
            // 16 K-chunks of 4 channels; one ds_load_b64 feeds A and B of two WMMAs.
            #pragma unroll
            for (int kc = 0; kc < 16; ++kc) {
                const int boff = kc * 4 + lhalf * 2;
                const float2 f = *(const float2*)(cur + lmod * LSTRIDE + boff);
                v2f a; a[0] = f.x; a[1] = f.y;
                // Gram: D[m,n] = sum_k X[m,k]*X[n,k]; same regs are valid A and B layouts.
                accQ = __builtin_amdgcn_wmma_f32_16x16x4_f32(
                    false, a, false, a, (short)0, accQ, false, false);
                // Row sums: D[m,n] = sum_k X[m,k] for every n.
                accS = __builtin_amdgcn_wmma_f32_16x16x4_f32(
                    false, a, false, ones, (short)0, accS, false, false);
            }
        }

        // C/D layout: VGPR r -> (lanes 0-15: M=r, N=lane) (lanes 16-31: M=8+r, N=lane-16).
        // Row sums (constant over N): lanes 0-15 hold rows 0-7, lanes 16-31 rows 8-15.
        // Gram diagonal row m: m<8 at lane m, vgpr m; m>=8 at lane 16+m (24..31), vgpr m-8.
        const bool act = (lane < 8) || (lane >= 24);
        if (act) {
            const int ci  = lane & 7;
            const int row = (lane < 8) ? lane : (lane - 16);
            const float rs = pick8(accS, ci);
            const float sq = pick8(accQ, ci);
            const int g = tgt[row0 + row];
            atomicAdd(&lsum[g], rs);
            atomicAdd(&lsq[g],  sq);
            atomicAdd(&lcnt[g], 1.0f);
        }
    }
    __syncthreads();
    for (int g = t; g < GRP; g += 192) {
        psum[blockIdx.x * GRP + g] = lsum[g];
        psq [blockIdx.x * GRP + g] = lsq[g];
        pcnt[blockIdx.x * GRP + g] = lcnt[g];
    }
}

// ---------------------------------------------------------------------------
// Kernel 2: reduce per-block partials -> proto (mean) / stdv per group.
// ---------------------------------------------------------------------------
__global__ void k2_finalize(const float* __restrict__ psum, const float* __restrict__ psq,
                            const float* __restrict__ pcnt,
                            float* __restrict__ proto, float* __restrict__ stdv,
                            double* __restrict__ acc)
{
    const int g = threadIdx.x;
    if (g == 0) *acc = 0.0;
    if (g < GRP) {
        float s = 0.f, q = 0.f, c = 0.f;
        for (int b = 0; b < NB1; ++b) {
            s += psum[b * GRP + g];
            q += psq [b * GRP + g];
            c += pcnt[b * GRP + g];
        }
        const float cnt  = c * (float)CCH;
        const float mean = s / cnt;
        float ss = q - cnt * mean * mean;
        ss = fmaxf(ss, 0.f);
        const float sd = (c > 1.5f) ? sqrtf(ss / fmaxf(cnt - 1.f, 1.f)) : 0.f;
        proto[g] = mean;
        stdv[g]  = sd;
    } else if (g == GRP) {
        proto[GRP] = 0.f;
        stdv[GRP]  = 0.f;
    }
}

// ---------------------------------------------------------------------------
// Kernel 3: masked mean numerator. One block per 'a'; sum over (b>a, x) pairs.
// ---------------------------------------------------------------------------
__global__ __launch_bounds__(256) void k3_vmean(const float* __restrict__ proto,
                                                const float* __restrict__ stdv,
                                                double* __restrict__ acc)
{
    __shared__ float sp[MM], sd[MM], qx[MM];
    __shared__ double red[256];
    const int t = threadIdx.x;
    const int a = blockIdx.x;

    for (int i = t; i < MM; i += 256) { sp[i] = proto[i]; sd[i] = stdv[i]; }
    __syncthreads();

    const float pa = sp[a];
    const float sa = sd[a];
    for (int i = t; i < MM; i += 256) {
        const float adx = fabsf(pa - sp[i]);
        qx[i] = adx / (adx * SQRT_C + EPSF);   // |d[x,a]|
    }
    __syncthreads();

    const int  nb    = (MM - 1) - a;            // b in (a, 383]
    const long total = (long)nb * MM;
    double lsum = 0.0;
    for (long i = t; i < total; i += 256) {
        const int b = a + 1 + (int)(i / MM);
        const int x = (int)(i % MM);
        const float sab = sa + sd[b];
        const float nab = fabsf(sp[b] - pa) * SQRT_C;
        const float s   = qx[x] * sab;
        const float v   = s / (nab + s + EPSF);
        lsum += (double)v;
    }
    red[t] = lsum;
    __syncthreads();
    for (int off = 128; off > 0; off >>= 1) {
        if (t < off) red[t] += red[t + off];
        __syncthreads();
    }
    if (t == 0) atomicAdd(acc, red[0]);         // global_atomic_add_f64, 384 total
}

// ---------------------------------------------------------------------------
// Kernel 4: emit mean = acc / M^3.
// ---------------------------------------------------------------------------
__global__ void k4_out(const double* __restrict__ acc, float* __restrict__ out)
{
    out[0] = (float)(acc[0] / 56623104.0);      // 384^3
}

extern "C" void kernel_launch(void* const* d_in, const int* in_sizes, int n_in,
                              void* d_out, int out_size, void* d_ws, size_t ws_size,
                              hipStream_t stream)
{
    const float* pred = (const float*)d_in[0];
    const int*   tgt  = (const int*)d_in[1];
    float*       out  = (float*)d_out;

    float* psum  = (float*)d_ws;                 // NB1*GRP
    float* psq   = psum + NB1 * GRP;             // NB1*GRP
    float* pcnt  = psq  + NB1 * GRP;             // NB1*GRP
    float* proto = pcnt + NB1 * GRP;             // MM
    float* stdv  = proto + MM;                   // MM
    // byte offset so far: (3*600*383 + 2*384)*4 = 2,760,672 (divisible by 8)
    double* acc  = (double*)(stdv + MM);

    k1_group_stats<<<NB1, 192, 0, stream>>>(pred, tgt, psum, psq, pcnt);
    k2_finalize  <<<1, 384, 0, stream>>>(psum, psq, pcnt, proto, stdv, acc);
    k3_vmean     <<<MM, 256, 0, stream>>>(proto, stdv, acc);
    k4_out       <<<1, 1, 0, stream>>>(acc, out);
}